// GraphSToV_9955734192541
// MI455X (gfx1250) — compile-verified
//
#include <hip/hip_runtime.h>

typedef __attribute__((ext_vector_type(2))) float v2f;
typedef __attribute__((ext_vector_type(4))) float f4;
typedef __attribute__((ext_vector_type(8))) float v8f;

#define FEAT    128   // ATOM_FEAT (F) == FILTERS (K) == 128
#define BN      1024  // B*N = 8*128
#define CDIM    3

// ---------------------------------------------------------------------------
// Kernel 1: dual GEMM via V_WMMA_F32_16X16X4_F32.
//   s1 = SF @ W[0:128, :],  s2 = SF @ W[128:256, :]
// One wave (32 threads) per 16x16 output tile position; each wave produces
// the tile for BOTH W-halves, reusing the A operand.
// ---------------------------------------------------------------------------
__global__ __launch_bounds__(32)
void GraphSToV_gemm_wmma(const float* __restrict__ SF,
                         const float* __restrict__ W,
                         float* __restrict__ s1,
                         float* __restrict__ s2) {
    const int lane  = threadIdx.x;          // 0..31, EXEC all ones
    const int m0    = blockIdx.x * 16;      // row tile in [0,1024)
    const int n0    = blockIdx.y * 16;      // col tile in [0,128)
    const int nlane = lane & 15;            // M row (A) / N col (B,C,D)
    const int khalf = (lane >> 4) << 1;     // A/B: lanes 16-31 hold K=2,3
    const int hi8   = (lane >> 4) << 3;     // C/D: lanes 16-31 hold M=r+8

    v8f acc1 = {0.f, 0.f, 0.f, 0.f, 0.f, 0.f, 0.f, 0.f};
    v8f acc2 = {0.f, 0.f, 0.f, 0.f, 0.f, 0.f, 0.f, 0.f};

    const float* Arow = SF + (size_t)(m0 + nlane) * FEAT;
    const float* W1   = W;                  // rows [0,128)
    const float* W2   = W + FEAT * FEAT;    // rows [128,256)

    for (int k = 0; k < FEAT; k += 4) {
        // A 16x4 f32: lane<16 -> {K=k,k+1}, lane>=16 -> {K=k+2,k+3}, M = lane&15
        v2f a;
        a.x = Arow[k + khalf];
        a.y = Arow[k + khalf + 1];
        // B 4x16 f32: lane<16 -> K rows {k,k+1}, lane>=16 -> {k+2,k+3}, N = lane&15
        const float* w1p = W1 + (size_t)(k + khalf) * FEAT + n0 + nlane;
        const float* w2p = W2 + (size_t)(k + khalf) * FEAT + n0 + nlane;
        v2f b1; b1.x = w1p[0]; b1.y = w1p[FEAT];
        v2f b2; b2.x = w2p[0]; b2.y = w2p[FEAT];
        // (neg_a, A, neg_b, B, c_mod, C, reuse_a, reuse_b)
        acc1 = __builtin_amdgcn_wmma_f32_16x16x4_f32(false, a, false, b1,
                                                     (short)0, acc1, false, false);
        acc2 = __builtin_amdgcn_wmma_f32_16x16x4_f32(false, a, false, b2,
                                                     (short)0, acc2, false, false);
    }

    // C/D 16x16 f32: VGPR r -> M = r (+8 for lanes 16-31), N = lane&15
#pragma unroll
    for (int r = 0; r < 8; ++r) {
        const size_t row = (size_t)(m0 + r + hi8);
        s1[row * FEAT + n0 + nlane] = acc1[r];
        s2[row * FEAT + n0 + nlane] = acc2[r];
    }
}

// ---------------------------------------------------------------------------
// Kernel 2: streaming epilogue (store-bandwidth bound, ~201 MB of writes).
//   out[b,i,j,c,k] = (s1[b,i,k] + s2[b,j,k] + bias[k]) * dist[b,i,j,c]
// One block per (b,i). s1 row + bias cached in LDS; 8 waves stride j;
// each lane owns a float4 of k; non-temporal stores for the write-once output.
// ---------------------------------------------------------------------------
__global__ __launch_bounds__(256)
void GraphSToV_epilogue(const float* __restrict__ s1,
                        const float* __restrict__ s2,
                        const float* __restrict__ bias,
                        const float* __restrict__ dist,
                        float* __restrict__ out) {
    const int bi   = blockIdx.x;        // b*128 + i, [0,1024)
    const int b    = bi >> 7;
    const int tid  = threadIdx.x;       // [0,256)
    const int lane = tid & 31;
    const int wave = tid >> 5;          // [0,8)

    __shared__ __align__(16) float s1p[FEAT];
    if (tid < FEAT) s1p[tid] = s1[(size_t)bi * FEAT + tid] + bias[tid];
    __syncthreads();

    const f4 s1v = *(const f4*)&s1p[lane * 4];

    const float* s2b = s2   + (size_t)b  * (128 * FEAT);
    const float* db  = dist + (size_t)bi * (128 * CDIM);
    float*       ob  = out  + (size_t)bi * (128 * CDIM * FEAT);

    for (int j = wave; j < 128; j += 8) {
        const f4 s2v = *(const f4*)&s2b[(size_t)j * FEAT + lane * 4];
        const f4 sum = s1v + s2v;
        const float d0 = db[j * CDIM + 0];
        const float d1 = db[j * CDIM + 1];
        const float d2 = db[j * CDIM + 2];
        float* o = ob + (size_t)j * (CDIM * FEAT) + lane * 4;
        __builtin_nontemporal_store(sum * d0, (f4*)(o));
        __builtin_nontemporal_store(sum * d1, (f4*)(o + FEAT));
        __builtin_nontemporal_store(sum * d2, (f4*)(o + 2 * FEAT));
    }
}

// ---------------------------------------------------------------------------
// inputs: [0] scalar_features (8,128,128) f32
//         [1] distances       (8,128,128,3) f32
//         [2] W               (256,128) f32
//         [3] b               (1,128) f32
// output: (8,128,128,3,128) f32 = 50,331,648 elements
// ws:     s1 (1024*128 f32) + s2 (1024*128 f32) = 1 MB
// ---------------------------------------------------------------------------
extern "C" void kernel_launch(void* const* d_in, const int* in_sizes, int n_in,
                              void* d_out, int out_size, void* d_ws, size_t ws_size,
                              hipStream_t stream) {
    const float* SF   = (const float*)d_in[0];
    const float* dist = (const float*)d_in[1];
    const float* W    = (const float*)d_in[2];
    const float* bias = (const float*)d_in[3];
    float* out = (float*)d_out;

    float* s1 = (float*)d_ws;
    float* s2 = s1 + (size_t)BN * FEAT;

    GraphSToV_gemm_wmma<<<dim3(BN / 16, FEAT / 16), 32, 0, stream>>>(SF, W, s1, s2);
    GraphSToV_epilogue<<<BN, 256, 0, stream>>>(s1, s2, bias, dist, out);
}